// My_loss_3118146257174
// MI455X (gfx1250) — compile-verified
//
#include <hip/hip_runtime.h>

// ---------------------------------------------------------------------------
// MI455X (gfx1250, wave32) fused physics-loss + sum reduction.
//
// Roofline: 128 MB of f32 input @ 23.3 TB/s ~= 5.5 us memory floor; ~40 VALU +
// 6 trans ops/element keeps compute overlapped. Strategy:
//   kernel 1: grid-stride float4 (global_load_b128) streaming, branchless
//             math (pow -> v_exp_f32(c*v_log_f32(x)), div -> v_rcp_f32),
//             per-thread f32 accumulator, per-block partial into d_ws.
//   kernel 2: single block folds the partials into d_out[0] (deterministic,
//             no float atomics).
// Cross-lane reduction uses the CDNA5 matrix engine: V_WMMA_F32_16X16X4_F32
// with A = per-lane partials, B = ones => D[m][n] = s(m) + s(m+16); summing
// the 8 D VGPRs per lane + one ds_swizzle swap-16 gives the full 32-lane sum.
// ---------------------------------------------------------------------------

typedef float v2f __attribute__((ext_vector_type(2)));
typedef float v8f __attribute__((ext_vector_type(8)));

#define THETA_W  1.9f
#define EPS_DPDT 1e-12f

// Hardware transcendentals (v_log_f32 is log2, v_exp_f32 is exp2).
__device__ __forceinline__ float hw_log2(float x) { return __builtin_amdgcn_logf(x); }
__device__ __forceinline__ float hw_exp2(float x) { return __builtin_amdgcn_exp2f(x); }
__device__ __forceinline__ float hw_rcp (float x) { return __builtin_amdgcn_rcpf(x); }

// Full 32-lane sum via the matrix engine (EXEC must be all ones here).
__device__ __forceinline__ float wave_sum32(float s) {
  v2f a; a.x = s;    a.y = 0.0f;   // lane L<16: A[L][0]=s ; lane L>=16: A[L-16][2]=s
  v2f b; b.x = 1.0f; b.y = 1.0f;   // B = ones(4x16)
  v8f c = {};
  c = __builtin_amdgcn_wmma_f32_16x16x4_f32(
      /*neg_a=*/false, a, /*neg_b=*/false, b,
      /*c_mod=*/(short)0, c, /*reuse_a=*/false, /*reuse_b=*/false);
  // lanes 0-15 hold rows 0..7 of D, lanes 16-31 hold rows 8..15; every column
  // of row m equals s(m)+s(m+16).
  float r = ((c[0] + c[1]) + (c[2] + c[3])) + ((c[4] + c[5]) + (c[6] + c[7]));
  // swap halves (lane ^ 16) with a single ds_swizzle_b32 offset:0x401F
  // (group-of-32 mode: xor_mask=0x10, or_mask=0, and_mask=0x1f).
  const int swz = __builtin_amdgcn_ds_swizzle(__float_as_int(r), 0x401f);
  r += __int_as_float(swz);        // total of all 32 lanes, in every lane
  return r;
}

// Block sum; valid result on threadIdx.x == 0.
__device__ __forceinline__ float block_sum(float s) {
  __shared__ float lds[32];
  float w = wave_sum32(s);
  const int lane = threadIdx.x & 31;
  const int wid  = threadIdx.x >> 5;
  if (lane == 0) lds[wid] = w;
  __syncthreads();
  float tot = 0.0f;
  if (threadIdx.x == 0) {
    const int nw = (blockDim.x + 31) >> 5;
    for (int i = 0; i < nw; ++i) tot += lds[i];
  }
  return tot;
}

struct Consts {
  float Vc, p_min, p_atm, p_crit;
  float invL, invL3, afr, inv_n, log2_patm;
  float beta_L, inv_betaL, beta_gain, e_pow, rho_mix_init;
};

__device__ __forceinline__ float elem_loss(float tgt, float dpdt, float mdot,
                                           float outP, const Consts& K) {
  const float p   = outP * 0.1f;
  const float pu  = fmaxf(p, K.p_min);
  const float x   = (pu - K.p_atm) * K.invL;
  const float th_mid = fmaf(x * x, fmaf(2.0f, x, -3.0f), 1.0f);   // 1 - 3x^2 + 2x^3
  const bool  lo  = (pu <= K.p_atm);
  const bool  hi  = (pu >= K.p_crit);
  const float theta  = lo ? 1.0f : (hi ? 0.0f : th_mid);
  const float dtheta = (lo || hi) ? 0.0f
                       : 6.0f * (pu - K.p_atm) * (pu - K.p_crit) * K.invL3;

  const float inv_pu = hw_rcp(pu);
  // (p_atm/pu)^(1/n) = exp2((log2(p_atm) - log2(pu)) / n)
  const float pr     = hw_exp2(K.inv_n * (K.log2_patm - hw_log2(pu)));
  const float afr_pr = K.afr * pr;
  const float p_denom = afr_pr * theta;
  const float p_ratio = afr_pr * fmaf(theta * inv_pu, K.inv_n, -dtheta);

  const float base = fmaf(K.beta_gain * (pu - K.p_atm), K.inv_betaL, 1.0f);
  // base^(-1 - 1/beta_gain) / beta_L
  const float exp_term = hw_exp2(K.e_pow * hw_log2(base)) * K.inv_betaL;

  const float denom = fmaf(K.beta_L * exp_term, base, p_denom);
  const float inv_d = hw_rcp(denom);
  const float drho  = K.rho_mix_init * (exp_term + p_ratio) * inv_d * inv_d;

  const float lp  = fabsf(fmaf(K.Vc * drho, dpdt, -mdot));
  const float lm  = fabsf(outP - tgt);
  const float per = fmaf(THETA_W, lp, lm);
  return (fabsf(dpdt) >= EPS_DPDT) ? per : 0.0f;
}

__global__ void fused_loss_partial(
    const float* __restrict__ tgtP,  const float* __restrict__ dpdtP,
    const float* __restrict__ mdotP, const float* __restrict__ Vp,
    const float* __restrict__ outPp, const float* __restrict__ rhoLp,
    const float* __restrict__ betaLp,const float* __restrict__ bgainp,
    const float* __restrict__ afp,   const float* __restrict__ rhogp,
    const float* __restrict__ polyp, const float* __restrict__ patmp,
    const float* __restrict__ pcritp,const float* __restrict__ pminp,
    float* __restrict__ partial, int n) {
  // Uniform scalar loads (L2-cached, once per thread), then derived constants.
  Consts K;
  K.Vc        = Vp[0];
  K.p_min     = pminp[0];
  K.p_atm     = patmp[0];
  K.p_crit    = pcritp[0];
  K.beta_L    = betaLp[0];
  K.beta_gain = bgainp[0];
  const float af    = afp[0];
  const float rhoL  = rhoLp[0];
  const float rhog  = rhogp[0];
  const float poly  = polyp[0];
  const float L     = K.p_crit - K.p_atm;
  K.invL         = 1.0f / L;
  K.invL3        = K.invL * K.invL * K.invL;
  K.afr          = af / (1.0f - af);
  K.inv_n        = 1.0f / poly;
  K.log2_patm    = hw_log2(K.p_atm);
  K.inv_betaL    = 1.0f / K.beta_L;
  K.e_pow        = -1.0f - 1.0f / K.beta_gain;
  K.rho_mix_init = fmaf(rhog, K.afr, rhoL);

  // 32-bit indexing: N <= 8M, n4 <= 2M — keeps the hot loop free of 64-bit
  // adds/compares and lets the addresser use SGPR-base + 32-bit voffset.
  const int tid    = (int)(blockIdx.x * blockDim.x + threadIdx.x);
  const int stride = (int)(gridDim.x * blockDim.x);
  const int n4     = n >> 2;

  const float4* __restrict__ t4 = reinterpret_cast<const float4*>(tgtP);
  const float4* __restrict__ d4 = reinterpret_cast<const float4*>(dpdtP);
  const float4* __restrict__ m4 = reinterpret_cast<const float4*>(mdotP);
  const float4* __restrict__ o4 = reinterpret_cast<const float4*>(outPp);

  float acc = 0.0f;
  for (int i = tid; i < n4; i += stride) {
    const float4 t = t4[i];
    const float4 d = d4[i];
    const float4 m = m4[i];
    const float4 o = o4[i];
    acc += elem_loss(t.x, d.x, m.x, o.x, K);
    acc += elem_loss(t.y, d.y, m.y, o.y, K);
    acc += elem_loss(t.z, d.z, m.z, o.z, K);
    acc += elem_loss(t.w, d.w, m.w, o.w, K);
  }
  // Scalar tail (n not multiple of 4).
  for (int i = n4 * 4 + tid; i < n; i += stride)
    acc += elem_loss(tgtP[i], dpdtP[i], mdotP[i], outPp[i], K);

  const float tot = block_sum(acc);   // all threads reach this: EXEC all ones
  if (threadIdx.x == 0) partial[blockIdx.x] = tot;
}

__global__ void reduce_partials(const float* __restrict__ partial,
                                float* __restrict__ out, int m) {
  float s = 0.0f;
  for (int i = threadIdx.x; i < m; i += blockDim.x) s += partial[i];
  const float tot = block_sum(s);
  if (threadIdx.x == 0) out[0] = tot;
}

extern "C" void kernel_launch(void* const* d_in, const int* in_sizes, int n_in,
                              void* d_out, int out_size, void* d_ws, size_t ws_size,
                              hipStream_t stream) {
  const int n = in_sizes[0];

  int nblocks = 2048;                               // ~524k threads, ~16 elem/thread
  const size_t need = (size_t)nblocks * sizeof(float);
  if (need > ws_size) nblocks = (int)(ws_size / sizeof(float));
  if (nblocks < 1) nblocks = 1;
  float* partial = reinterpret_cast<float*>(d_ws);

  fused_loss_partial<<<nblocks, 256, 0, stream>>>(
      (const float*)d_in[0],  (const float*)d_in[1],  (const float*)d_in[2],
      (const float*)d_in[3],  (const float*)d_in[4],  (const float*)d_in[5],
      (const float*)d_in[6],  (const float*)d_in[7],  (const float*)d_in[8],
      (const float*)d_in[9],  (const float*)d_in[10], (const float*)d_in[11],
      (const float*)d_in[12], (const float*)d_in[13],
      partial, n);

  reduce_partials<<<1, 256, 0, stream>>>(partial, (float*)d_out, nblocks);
}